// TopologicalTrainer_91182155694875
// MI455X (gfx1250) — compile-verified
//
#include <hip/hip_runtime.h>

// ---------------------------------------------------------------------------
// Sheaf-Laplacian spectral-gap triplet loss on gfx1250 (MI455X, wave32).
//   Per cloud (sequential, so the 64MB Laplacian stays resident in 192MB L2):
//     X = [q;p] @ W^T                     (WMMA f32 16x16x4 GEMM)
//     KNN(5) graph, restriction maps, assemble L (4096x4096 f32)
//     Blocked Householder tridiagonalization (LATRD/SYTRD style):
//        - per column: SYMV p = A*v (wave/row, L2-resident) + panel-corrected
//          Householder in block 0 (2 device-wide syncs per column)
//        - per 32-col panel: SYR2K  A -= V*W^T + W*V^T  as K=64 WMMA GEMM
//     Sturm bisection eigenvalues (1 thread / eigenvalue)
//     soft-gated spectral gap; final relu(gp-gn+0.5)+0.1*gp
// ---------------------------------------------------------------------------

typedef __attribute__((ext_vector_type(2))) float v2f;
typedef __attribute__((ext_vector_type(8))) float v8f;

#define NPTS   512
#define DIN    768
#define SDIM   8
#define NMAT   4096           // NPTS * SDIM
#define KNN_K  5
#define NB     32             // tridiag panel width (K = 2*NB = 64 for SYR2K)
#define TB_BLOCKS  120        // persistent grid (must be co-resident)
#define TB_THREADS 256

// ------------------------------ utilities ----------------------------------

__global__ void zero_f32_kernel(float* __restrict__ p, int nfl) {
  for (int t = blockIdx.x * blockDim.x + threadIdx.x; t < nfl;
       t += gridDim.x * blockDim.x)
    p[t] = 0.0f;
}

__global__ void init_barrier_kernel(unsigned* b) {
  if (threadIdx.x == 0 && blockIdx.x == 0) { b[0] = 0u; b[1] = 0u; }
}

__device__ __forceinline__ void grid_sync(unsigned* cnt, unsigned* gen,
                                          int nblocks) {
  __syncthreads();
  if (threadIdx.x == 0) {
    __threadfence();
    volatile unsigned* vgen = (volatile unsigned*)gen;
    unsigned g = *vgen;
    unsigned arrived = atomicAdd(cnt, 1u);
    if (arrived == (unsigned)(nblocks - 1)) {
      *(volatile unsigned*)cnt = 0u;
      __threadfence();
      atomicAdd(gen, 1u);
    } else {
      while (*vgen == g) { __builtin_amdgcn_s_sleep(2); }
      __threadfence();
    }
  }
  __syncthreads();
}

// --------------------- stage A: projection GEMM (WMMA) ---------------------
// X[i][j] = dot(points[i], W[j]); unconditional loads + cndmask (no execz
// branching around the B-operand loads).

__global__ __launch_bounds__(32) void proj_wmma_kernel(
    const float* __restrict__ q, const float* __restrict__ other,
    const float* __restrict__ W, float* __restrict__ X) {
  const int tile = blockIdx.x;          // 32 tiles of 16 rows
  const int lane = threadIdx.x;         // wave32
  const int hi = lane >> 4;             // K offset 0 or 2
  const int lo = lane & 15;             // M (A) / N (B,C)
  const int row0 = tile * 16;

  v8f acc;
  for (int r = 0; r < 8; ++r) acc[r] = 0.0f;

  const int arow = row0 + lo;
  const float* aptr =
      (arow < 256) ? (q + (size_t)arow * DIN) : (other + (size_t)(arow - 256) * DIN);
  const bool bvalid = (lo < SDIM);
  const float* bptr = W + (size_t)(lo & (SDIM - 1)) * DIN;   // always valid

  for (int k0 = 0; k0 < DIN; k0 += 4) {
    const int ka = k0 + hi * 2;
    v2f a;  a.x = aptr[ka];  a.y = aptr[ka + 1];
    const float bx = bptr[ka];
    const float by = bptr[ka + 1];
    v2f b;  b.x = bvalid ? bx : 0.0f;  b.y = bvalid ? by : 0.0f;
    acc = __builtin_amdgcn_wmma_f32_16x16x4_f32(false, a, false, b, (short)0,
                                                acc, false, false);
  }
  for (int r = 0; r < 8; ++r) {
    const int row = row0 + r + 8 * hi;
    if (lo < SDIM) X[row * SDIM + lo] = acc[r];
  }
}

// --------------------- stage B: pairwise distances -------------------------

__global__ void pairdist_kernel(const float* __restrict__ X,
                                float* __restrict__ D2) {
  int t = blockIdx.x * blockDim.x + threadIdx.x;
  if (t >= NPTS * NPTS) return;
  int i = t >> 9, j = t & (NPTS - 1);
  float s = 0.0f;
  for (int c = 0; c < SDIM; ++c) {
    float d = X[i * SDIM + c] - X[j * SDIM + c];
    s += d * d;
  }
  D2[t] = s;
}

// --------------------- stage C: KNN + symmetrize ---------------------------

__global__ void knn_kernel(const float* __restrict__ D2,
                           unsigned char* __restrict__ adjD) {
  int i = blockIdx.x * blockDim.x + threadIdx.x;
  if (i >= NPTS) return;
  int chosen[KNN_K];
  for (int t = 0; t < KNN_K; ++t) {
    float bd = 3.4e38f; int bj = -1;
    for (int j = 0; j < NPTS; ++j) {
      if (j == i) continue;
      bool used = false;
      for (int u = 0; u < t; ++u) used = used || (chosen[u] == j);
      if (used) continue;
      float d = D2[i * NPTS + j];
      if (d < bd) { bd = d; bj = j; }
    }
    chosen[t] = bj;
    adjD[i * NPTS + bj] = 1;
  }
}

__global__ void sym_kernel(const unsigned char* __restrict__ adjD,
                           unsigned char* __restrict__ adj) {
  int t = blockIdx.x * blockDim.x + threadIdx.x;
  if (t >= NPTS * NPTS) return;
  int i = t >> 9, j = t & (NPTS - 1);
  adj[t] = (unsigned char)(adjD[t] | adjD[j * NPTS + i]);
}

// --------------------- stage D: assemble L ---------------------------------

__global__ __launch_bounds__(64) void diag_kernel(
    const float* __restrict__ X, const unsigned char* __restrict__ adj,
    float* __restrict__ L) {
  const int t = blockIdx.x;
  const int a = threadIdx.x >> 3, b = threadIdx.x & 7;
  __shared__ float sd[SDIM];
  __shared__ float scoef;
  float acc = 0.0f;
  for (int i = 0; i < t; ++i) {
    if (!adj[i * NPTS + t]) continue;          // block-uniform condition
    if (threadIdx.x < SDIM)
      sd[threadIdx.x] = X[t * SDIM + threadIdx.x] - X[i * SDIM + threadIdx.x];
    __syncthreads();
    if (threadIdx.x == 0) {
      float n2 = 0.0f;
      for (int c = 0; c < SDIM; ++c) n2 += sd[c] * sd[c];
      float dn = sqrtf(n2) + 1e-12f;
      float alpha = fminf(dn, 1.0f);
      scoef = alpha * (2.0f - alpha) / (dn * dn);
    }
    __syncthreads();
    acc += ((a == b) ? 1.0f : 0.0f) - scoef * sd[a] * sd[b];
    __syncthreads();
  }
  int deg = 0;
  for (int j = t + 1; j < NPTS; ++j) deg += adj[t * NPTS + j];
  if (a == b) acc += (float)deg;
  L[(size_t)(t * SDIM + a) * NMAT + (t * SDIM + b)] = acc;
}

__global__ void offdiag_kernel(const float* __restrict__ X,
                               const unsigned char* __restrict__ adj,
                               float* __restrict__ L) {
  int t = blockIdx.x * blockDim.x + threadIdx.x;
  if (t >= NPTS * NPTS) return;
  int i = t >> 9, j = t & (NPTS - 1);
  if (i >= j || !adj[t]) return;
  float d[SDIM]; float n2 = 0.0f;
  for (int c = 0; c < SDIM; ++c) {
    d[c] = X[j * SDIM + c] - X[i * SDIM + c];
    n2 += d[c] * d[c];
  }
  float dn = sqrtf(n2) + 1e-12f;
  float alpha = fminf(dn, 1.0f);
  float coef = alpha / (dn * dn);
  for (int a = 0; a < SDIM; ++a)
    for (int b = 0; b < SDIM; ++b) {
      float r = ((a == b) ? 1.0f : 0.0f) - coef * d[a] * d[b];
      L[(size_t)(i * SDIM + a) * NMAT + (j * SDIM + b)] = -r;  // -R
      L[(size_t)(j * SDIM + a) * NMAT + (i * SDIM + b)] = -r;  // -R^T = -R
    }
}

// --------------------- stage E: blocked Householder tridiagonalization -----
// LATRD-style: per column j of a 32-wide panel (k = k0+j):
//   block 0:  finish w_{j-1};  a_col -= V*Wrow^T + W*Vrow^T;  Householder
//             (tau, alpha, v_j);  c1 = W^T v, c2 = V^T v   | grid sync
//   grid   :  praw = A_stored * v  (wave per row, A is L2-resident) + partial
//             v.praw                                        | grid sync
// per panel: finish w_{nb-1} (grid), then SYR2K
//             A(t0:,t0:) -= V*W^T + W*V^T  via K=64 chained WMMA  (2 syncs)

__device__ __forceinline__ float panel_elem(const float* __restrict__ P0,
                                            const float* __restrict__ P1,
                                            int u, int g, int nb, int n) {
  if (g >= n) return 0.0f;
  if (u < nb) return P0[(size_t)u * n + g];
  if (u < 2 * nb) return P1[(size_t)(u - nb) * n + g];
  return 0.0f;
}

__global__ __launch_bounds__(TB_THREADS) void tridiag_kernel(
    float* __restrict__ A, float* __restrict__ Vb, float* __restrict__ Wb,
    float* __restrict__ pbuf, float* __restrict__ dbuf,
    float* __restrict__ ebuf, float* __restrict__ partials,
    float* __restrict__ scalars, unsigned* __restrict__ bar) {
  const int n = NMAT;
  const int tid = threadIdx.x;
  const int nthreads = blockDim.x;
  const int nblocks = gridDim.x;
  const int lane = tid & 31;
  const int wpb = nthreads >> 5;
  const int gwave = blockIdx.x * wpb + (tid >> 5);
  const int nwaves = nblocks * wpb;

  __shared__ float sv[NMAT];          // current Householder vector
  __shared__ float red[TB_THREADS];
  __shared__ float sVrow[NB], sWrow[NB];
  __shared__ float sc1[NB], sc2[NB];  // block-0 private panel scalars
  __shared__ float sTau, sDotvp, sCC;

  unsigned* cnt = bar;
  unsigned* gen = bar + 1;

  for (int k0 = 0; k0 < n - 2; k0 += NB) {
    const int nbEff = min(NB, n - 2 - k0);

    for (int j = 0; j < nbEff; ++j) {
      const int k = k0 + j;
      const int base = k + 1;

      // ---------------- phase A (block 0 only) ----------------
      if (blockIdx.x == 0) {
        if (j > 0) {
          // finish w_{j-1}: w = tau*(p - V c1 - W c2) - 0.5 tau (v^T w) v
          if (tid == 0) {
            float dv = 0.0f;
            for (int u = 0; u < nwaves; ++u) dv += partials[u];
            float cc = 0.0f;
            for (int u = 0; u < j - 1; ++u) cc += sc1[u] * sc2[u];
            sDotvp = dv; sCC = cc;
          }
          __syncthreads();
          const float tauP = sTau;
          const float wv = tauP * (sDotvp - 2.0f * sCC);
          const float* vcol = Vb + (size_t)(j - 1) * n;
          float* wcol = Wb + (size_t)(j - 1) * n;
          for (int g = tid; g < n; g += nthreads) {
            float w = 0.0f;
            if (g >= k) {                       // support of column j-1
              float z = pbuf[g];
              for (int u = 0; u < j - 1; ++u)
                z -= Vb[(size_t)u * n + g] * sc1[u] +
                     Wb[(size_t)u * n + g] * sc2[u];
              w = tauP * z - 0.5f * tauP * wv * vcol[g];
            }
            wcol[g] = w;
          }
          __syncthreads();
        }
        // panel row k
        if (tid < j) {
          sVrow[tid] = Vb[(size_t)tid * n + k];
          sWrow[tid] = Wb[(size_t)tid * n + k];
        }
        __syncthreads();
        // a_col update + sigma^2 (result kept only in V column j)
        float* vj = Vb + (size_t)j * n;
        float part = 0.0f;
        for (int g = base + tid; g < n; g += nthreads) {
          float x = A[(size_t)g * n + k];
          for (int u = 0; u < j; ++u)
            x -= Vb[(size_t)u * n + g] * sWrow[u] +
                 Wb[(size_t)u * n + g] * sVrow[u];
          vj[g] = x;
          part += x * x;
        }
        red[tid] = part;
        __syncthreads();
        if (tid == 0) {
          float sig2 = 0.0f;
          for (int u = 0; u < nthreads; ++u) sig2 += red[u];
          float x0 = vj[base];
          float sig = sqrtf(sig2);
          float alpha = (x0 >= 0.0f) ? -sig : sig;
          float v0 = x0 - alpha;
          float vtv = sig2 - x0 * x0 + v0 * v0;
          float tau = (vtv > 1e-30f) ? (2.0f / vtv) : 0.0f;
          vj[base] = v0;
          float dk = A[(size_t)k * n + k];
          for (int u = 0; u < j; ++u) dk -= 2.0f * sVrow[u] * sWrow[u];
          dbuf[k] = dk;
          ebuf[k] = alpha;
          sTau = tau;
          scalars[2 * NB] = tau;
        }
        __syncthreads();
        for (int g = tid; g <= k; g += nthreads) vj[g] = 0.0f;
        __syncthreads();
        // c1 = W^T v, c2 = V^T v (fixed-order block reductions)
        for (int u = 0; u < j; ++u) {
          float p1 = 0.0f, p2 = 0.0f;
          for (int g = base + tid; g < n; g += nthreads) {
            float vg = vj[g];
            p1 += Wb[(size_t)u * n + g] * vg;
            p2 += Vb[(size_t)u * n + g] * vg;
          }
          red[tid] = p1; __syncthreads();
          if (tid == 0) {
            float s = 0.0f;
            for (int t2 = 0; t2 < nthreads; ++t2) s += red[t2];
            sc1[u] = s; scalars[u] = s;
          }
          __syncthreads();
          red[tid] = p2; __syncthreads();
          if (tid == 0) {
            float s = 0.0f;
            for (int t2 = 0; t2 < nthreads; ++t2) s += red[t2];
            sc2[u] = s; scalars[NB + u] = s;
          }
          __syncthreads();
        }
      }
      grid_sync(cnt, gen, nblocks);

      // ---------------- phase B (grid): praw = A * v ----------------
      for (int g = base + tid; g < n; g += nthreads) sv[g] = Vb[(size_t)j * n + g];
      __syncthreads();

      float pvp = 0.0f;
      const int m = n - base;
      for (int r = gwave; r < m; r += nwaves) {
        const int g = base + r;
        const float* row = A + (size_t)g * n;
        if (r + nwaves < m)
          __builtin_prefetch(A + (size_t)(g + nwaves) * n + base + lane, 0, 1);
        float s = 0.0f;
        for (int h = base + lane; h < n; h += 32) s += row[h] * sv[h];
        for (int off = 16; off > 0; off >>= 1) s += __shfl_xor(s, off, 32);
        if (lane == 0) { pbuf[g] = s; pvp += sv[g] * s; }
      }
      if (lane == 0) partials[gwave] = pvp;
      grid_sync(cnt, gen, nblocks);
    }

    // ---------------- finish w_{nbEff-1} (whole grid) ----------------
    {
      const int jl = nbEff - 1;
      const int bp = k0 + nbEff;            // support start of last column
      if (tid == 0) {
        float dv = 0.0f;
        for (int u = 0; u < nwaves; ++u) dv += partials[u];
        float cc = 0.0f;
        for (int u = 0; u < jl; ++u) cc += scalars[u] * scalars[NB + u];
        sDotvp = dv; sCC = cc;
      }
      __syncthreads();
      const float tauL = scalars[2 * NB];
      const float wv = tauL * (sDotvp - 2.0f * sCC);
      const float* vcol = Vb + (size_t)jl * n;
      float* wcol = Wb + (size_t)jl * n;
      for (int g = blockIdx.x * nthreads + tid; g < n; g += nblocks * nthreads) {
        float w = 0.0f;
        if (g >= bp) {
          float z = pbuf[g];
          for (int u = 0; u < jl; ++u)
            z -= Vb[(size_t)u * n + g] * scalars[u] +
                 Wb[(size_t)u * n + g] * scalars[NB + u];
          w = tauL * z - 0.5f * tauL * wv * vcol[g];
        }
        wcol[g] = w;
      }
    }
    grid_sync(cnt, gen, nblocks);

    // ---------------- SYR2K:  A(t0:,t0:) -= V W^T + W V^T  (WMMA) --------
    {
      const int t0 = k0 + nbEff;
      const int mrem = n - t0;
      if (mrem > 0) {
        const int mt = (mrem + 15) >> 4;
        const int ntiles = mt * mt;
        const int kiters = (2 * nbEff + 3) >> 2;   // K = 2*nbEff in steps of 4
        const int hi = lane >> 4;
        const int lo = lane & 15;
        for (int t = gwave; t < ntiles; t += nwaves) {
          const int rb = t0 + (t / mt) * 16;
          const int cb = t0 + (t % mt) * 16;
          const int ra = rb + lo;
          const int ca = cb + lo;
          v8f c;
          for (int r = 0; r < 8; ++r) {
            const int row = rb + r + 8 * hi;
            c[r] = (row < n && ca < n) ? A[(size_t)row * n + ca] : 0.0f;
          }
          for (int kk = 0; kk < kiters; ++kk) {
            const int u0 = kk * 4 + hi * 2;
            v2f aop, bop;
            aop.x = -panel_elem(Vb, Wb, u0,     ra, nbEff, n);   // U = [-V -W]
            aop.y = -panel_elem(Vb, Wb, u0 + 1, ra, nbEff, n);
            bop.x =  panel_elem(Wb, Vb, u0,     ca, nbEff, n);   // B = [W; V]
            bop.y =  panel_elem(Wb, Vb, u0 + 1, ca, nbEff, n);
            c = __builtin_amdgcn_wmma_f32_16x16x4_f32(false, aop, false, bop,
                                                      (short)0, c, false, false);
          }
          for (int r = 0; r < 8; ++r) {
            const int row = rb + r + 8 * hi;
            if (row < n && ca < n) A[(size_t)row * n + ca] = c[r];
          }
        }
      }
    }
    grid_sync(cnt, gen, nblocks);
  }

  if (blockIdx.x == 0 && tid == 0) {
    dbuf[n - 2] = A[(size_t)(n - 2) * n + (n - 2)];
    ebuf[n - 2] = A[(size_t)(n - 1) * n + (n - 2)];
    dbuf[n - 1] = A[(size_t)(n - 1) * n + (n - 1)];
  }
}

// --------------------- stage F: Sturm bisection ----------------------------

__global__ __launch_bounds__(256) void bisect_kernel(
    const float* __restrict__ dbuf, const float* __restrict__ ebuf,
    float* __restrict__ eigs) {
  const int n = NMAT;
  __shared__ float sd[NMAT];
  __shared__ float se2[NMAT];
  __shared__ float sbounds[2];
  for (int g = threadIdx.x; g < n; g += blockDim.x) {
    sd[g] = dbuf[g];
    float e = (g < n - 1) ? ebuf[g] : 0.0f;
    se2[g] = e * e;
  }
  __syncthreads();
  if (threadIdx.x == 0) {
    float lo = 3.4e38f, hi = -3.4e38f;
    for (int g = 0; g < n; ++g) {
      float r = ((g > 0) ? sqrtf(se2[g - 1]) : 0.0f) +
                ((g < n - 1) ? sqrtf(se2[g]) : 0.0f);
      lo = fminf(lo, sd[g] - r);
      hi = fmaxf(hi, sd[g] + r);
    }
    sbounds[0] = lo - 1e-3f;
    sbounds[1] = hi + 1e-3f;
  }
  __syncthreads();
  const int idx = blockIdx.x * blockDim.x + threadIdx.x;
  if (idx >= n) return;
  float lo = sbounds[0], hi = sbounds[1];
  for (int it = 0; it < 60; ++it) {
    float mid = 0.5f * (lo + hi);
    int cntneg = 0;
    float qv = 1.0f;
    for (int g = 0; g < n; ++g) {
      float denom = qv;
      if (fabsf(denom) < 1e-30f) denom = (denom < 0.0f) ? -1e-30f : 1e-30f;
      qv = (sd[g] - mid) - ((g > 0) ? se2[g - 1] / denom : 0.0f);
      if (qv < 0.0f) cntneg++;
    }
    if (cntneg > idx) hi = mid; else lo = mid;
  }
  eigs[idx] = fmaxf(0.5f * (lo + hi), 0.0f);   // clip(eigs, 0)
}

// --------------------- stage G: gap + combine ------------------------------

__global__ __launch_bounds__(256) void gap_kernel(const float* __restrict__ eigs,
                                                  float* __restrict__ gaps,
                                                  int which) {
  __shared__ float rn[256], rd[256];
  float sn = 0.0f, sdn = 0.0f;
  for (int g = threadIdx.x; g < NMAT; g += 256) {
    float e = eigs[g];
    float gate = 1.0f / (1.0f + expf(-(e - 1e-4f) / 0.01f));
    sn += e * gate;
    sdn += gate;
  }
  rn[threadIdx.x] = sn;
  rd[threadIdx.x] = sdn;
  __syncthreads();
  if (threadIdx.x == 0) {
    float a = 0.0f, b = 0.0f;
    for (int u = 0; u < 256; ++u) { a += rn[u]; b += rd[u]; }
    gaps[which] = a / (b + 1e-12f);
  }
}

__global__ void final_kernel(const float* __restrict__ gaps,
                             float* __restrict__ out) {
  if (threadIdx.x == 0 && blockIdx.x == 0) {
    float gp = gaps[0], gn = gaps[1];
    out[0] = fmaxf(gp - gn + 0.5f, 0.0f) + 0.1f * gp;
  }
}

// --------------------------- host orchestration ----------------------------

extern "C" void kernel_launch(void* const* d_in, const int* in_sizes, int n_in,
                              void* d_out, int out_size, void* d_ws,
                              size_t ws_size, hipStream_t stream) {
  (void)in_sizes; (void)n_in; (void)out_size; (void)ws_size;
  const float* q   = (const float*)d_in[0];
  const float* pos = (const float*)d_in[1];
  const float* neg = (const float*)d_in[2];
  const float* W   = (const float*)d_in[3];

  char* ws = (char*)d_ws;
  size_t off = 0;
  auto alloc = [&](size_t bytes) -> char* {
    char* p = ws + off;
    off += (bytes + 255) & ~(size_t)255;
    return p;
  };
  float*         L        = (float*)alloc(sizeof(float) * (size_t)NMAT * NMAT);
  float*         Vb       = (float*)alloc(sizeof(float) * (size_t)NMAT * NB);
  float*         Wb       = (float*)alloc(sizeof(float) * (size_t)NMAT * NB);
  float*         X        = (float*)alloc(sizeof(float) * NPTS * SDIM);
  float*         D2       = (float*)alloc(sizeof(float) * NPTS * NPTS);
  unsigned char* adjD     = (unsigned char*)alloc(NPTS * NPTS);
  unsigned char* adj      = (unsigned char*)alloc(NPTS * NPTS);
  float*         pbuf     = (float*)alloc(sizeof(float) * NMAT);
  float*         dbuf     = (float*)alloc(sizeof(float) * NMAT);
  float*         ebuf     = (float*)alloc(sizeof(float) * NMAT);
  float*         eigs     = (float*)alloc(sizeof(float) * NMAT);
  float*         partials = (float*)alloc(sizeof(float) * 1024);
  float*         scalars  = (float*)alloc(sizeof(float) * 128);
  unsigned*      bar      = (unsigned*)alloc(256);
  float*         gaps     = (float*)alloc(sizeof(float) * 2);

  for (int cloud = 0; cloud < 2; ++cloud) {
    const float* other = (cloud == 0) ? pos : neg;
    proj_wmma_kernel<<<NPTS / 16, 32, 0, stream>>>(q, other, W, X);
    pairdist_kernel<<<(NPTS * NPTS + 255) / 256, 256, 0, stream>>>(X, D2);
    zero_f32_kernel<<<256, 256, 0, stream>>>((float*)adjD, NPTS * NPTS / 4);
    knn_kernel<<<(NPTS + 255) / 256, 256, 0, stream>>>(D2, adjD);
    sym_kernel<<<(NPTS * NPTS + 255) / 256, 256, 0, stream>>>(adjD, adj);
    zero_f32_kernel<<<2048, 256, 0, stream>>>(L, NMAT * NMAT);
    diag_kernel<<<NPTS, 64, 0, stream>>>(X, adj, L);
    offdiag_kernel<<<(NPTS * NPTS + 255) / 256, 256, 0, stream>>>(X, adj, L);
    init_barrier_kernel<<<1, 32, 0, stream>>>(bar);
    tridiag_kernel<<<TB_BLOCKS, TB_THREADS, 0, stream>>>(
        L, Vb, Wb, pbuf, dbuf, ebuf, partials, scalars, bar);
    bisect_kernel<<<NMAT / 256, 256, 0, stream>>>(dbuf, ebuf, eigs);
    gap_kernel<<<1, 256, 0, stream>>>(eigs, gaps, cloud);
  }
  final_kernel<<<1, 32, 0, stream>>>(gaps, (float*)d_out);
}